// RationalQuadraticSpline_19971597926503
// MI455X (gfx1250) — compile-verified
//
#include <hip/hip_runtime.h>
#include <cstdint>

// Rational-quadratic spline flow, N = 4.19M elements, params row = 25 f32.
// HBM-bound: ~453 MB traffic -> ~19.5 us floor at 23.3 TB/s; no matrix ops
// exist in the workload, so the CDNA5-specific lever is data movement:
//   - one TENSOR_LOAD_TO_LDS (TDM) per block stages the 25,600 B param tile
//     into LDS (descriptor built in SGPRs by wave 0, S_WAIT_TENSORCNT drain),
//   - per-lane global_load_async_to_lds_b32 only for the tail block,
//   - NT loads/stores for streamed-once x/out.

#define NBINS   8
#define TAILB   3.0f
#define MIN_BW  0.001f
#define MIN_BH  0.001f
#define MIN_D   0.001f
#define EPSV    1e-6f

#define THREADS 256
#define ROW     25                         // 3*NBINS + 1
#define FLOATS_PER_BLOCK (THREADS * ROW)   // 6400 floats = 25600 B LDS tile

typedef unsigned int v4u __attribute__((ext_vector_type(4)));
typedef unsigned int v8u __attribute__((ext_vector_type(8)));

// Low 32 bits of a generic pointer to __shared__ data == LDS byte offset
// (flat-address aperture scheme: LDS offset is the low 32 address bits).
__device__ __forceinline__ unsigned lds_off_u32(const void* p) {
    return (unsigned)(unsigned long long)p;
}

__device__ __forceinline__ void async_ld_b32(unsigned lds_byte, const void* gaddr) {
    asm volatile("global_load_async_to_lds_b32 %0, %1, off"
                 :: "v"(lds_byte), "v"(gaddr) : "memory");
}
__device__ __forceinline__ void wait_async0() {
    asm volatile("s_wait_asynccnt 0" ::: "memory");
}

// One TDM transfer: contiguous tile of `nelem` f32 from gaddr to the LDS byte offset in lds_byte.
// D# per CDNA5 ISA: group0 {count=1, lds_addr,
// global_addr, type=2}, group1 {data_size=4B, tensor_dim0=nelem,
// tensor_dim1=1, tile_dim0=nelem, tile_dim1=1, stride0=nelem}.
__device__ __forceinline__ void tdm_load_f32_tile(unsigned lds_byte,
                                                  const void* gaddr,
                                                  unsigned nelem) {
    unsigned long long ga = (unsigned long long)gaddr;
    v4u g0;
    g0[0] = 1u;                                   // count=1, user mode
    g0[1] = lds_byte;                             // lds_addr
    g0[2] = (unsigned)ga;                         // global_addr[31:0]
    g0[3] = (unsigned)(ga >> 32) | (2u << 30);    // global_addr[56:32] | type=2

    v8u g1;
    g1[0] = 2u << 16;                             // workgroup_mask=0, data_size=4B
    g1[1] = (nelem & 0xFFFFu) << 16;              // tensor_dim0[15:0]
    g1[2] = (nelem >> 16) | (1u << 16);           // tensor_dim0[31:16] | tensor_dim1=1
    g1[3] = (nelem & 0xFFFFu) << 16;              // tile_dim0 (16-bit)
    g1[4] = 1u;                                   // tile_dim1=1, tile_dim2=0
    g1[5] = nelem;                                // tensor_dim0_stride[31:0]
    g1[6] = 0u;                                   // stride0[47:32] | stride1[15:0]
    g1[7] = 0u;                                   // stride1[47:16]

    asm volatile("tensor_load_to_lds %0, %1" :: "s"(g0), "s"(g1) : "memory");
}

__device__ __forceinline__ float fast_softplus(float v) {
    // log(1+exp(v)) = max(v,0) + log(1+exp(-|v|)) : stable, hw-trans based
    float a = __builtin_fabsf(v);
    return fmaxf(v, 0.0f) + __logf(1.0f + __expf(-a));
}

// Replicates _cum_knots: softmax*2B, rescale by (2B - K*min) and add min,
// cumsum, prepend -B, force last knot to +B. c has NBINS+1 entries.
__device__ __forceinline__ void cum_knots(const float* u, float min_size, float* c) {
    float m = u[0];
#pragma unroll
    for (int k = 1; k < NBINS; ++k) m = fmaxf(m, u[k]);
    float e[NBINS];
    float s = 0.0f;
#pragma unroll
    for (int k = 0; k < NBINS; ++k) { e[k] = __expf(u[k] - m); s += e[k]; }
    const float scale = (2.0f * TAILB) * (2.0f * TAILB - (float)NBINS * min_size) / s;
    float run = 0.0f;
    c[0] = -TAILB;
#pragma unroll
    for (int k = 0; k < NBINS; ++k) {
        float w = e[k] * scale + min_size;
        run += w;
        c[k + 1] = run;
    }
    c[NBINS] = TAILB; // .at[..., -1].set(B)
}

__global__ __launch_bounds__(THREADS)
void rqs_kernel(const float* __restrict__ x, const float* __restrict__ params,
                float* __restrict__ out, float* __restrict__ partial, int n)
{
    __shared__ __align__(16) float sp[FLOATS_PER_BLOCK];
    __shared__ float sred[THREADS / 32];

    const int tid = threadIdx.x;
    const long long base = (long long)blockIdx.x * THREADS;
    long long rem = (long long)n - base;
    const int rows = rem >= THREADS ? THREADS : (int)rem;
    const float* prow = params + base * ROW;
    const bool full = (rows == THREADS);

    // ---- Stage this block's param rows into LDS ----
    if (full) {
        if (tid < 32) {  // producer wave issues one TDM DMA (EXEC=all-1s in wave 0)
            tdm_load_f32_tile(lds_off_u32(&sp[0]), (const void*)prow,
                              (unsigned)FLOATS_PER_BLOCK);
        }
    } else {             // safety tail (not hit for N = 4M): per-lane async copies
        const int nf = rows * ROW;
        for (int j = tid; j < nf; j += THREADS)
            async_ld_b32(lds_off_u32(&sp[j]), (const void*)(prow + j));
    }

    // Overlap the x load with the LDS staging.
    float xv = 0.0f;
    if (tid < rows) xv = __builtin_nontemporal_load(&x[base + tid]);

    if (full) {
        if (tid < 32) __builtin_amdgcn_s_wait_tensorcnt(0);
    } else {
        wait_async0();
    }
    __syncthreads();     // make LDS visible across the 8 waves

    float lad = 0.0f;
    if (tid < rows) {
        const float* row = &sp[tid * ROW]; // stride-25 LDS reads: conflict-free
        float uw[NBINS], uh[NBINS], ud[NBINS + 1];
#pragma unroll
        for (int k = 0; k < NBINS; ++k) uw[k] = row[k];
#pragma unroll
        for (int k = 0; k < NBINS; ++k) uh[k] = row[NBINS + k];
#pragma unroll
        for (int k = 0; k <= NBINS; ++k) ud[k] = row[2 * NBINS + k];

        float cw[NBINS + 1], ch[NBINS + 1], dv[NBINS + 1];
        cum_knots(uw, MIN_BW, cw);
        cum_knots(uh, MIN_BH, ch);
#pragma unroll
        for (int k = 0; k <= NBINS; ++k) dv[k] = fast_softplus(ud[k]) + MIN_D;

        const bool inside = (xv >= -TAILB) && (xv <= TAILB);
        const float xi = fminf(fmaxf(xv, -TAILB), TAILB);

        // idx = sum(xi >= bin_loc) - 1, bin_loc = cw with last += EPS
        int idx = -1;
#pragma unroll
        for (int j = 0; j <= NBINS; ++j) {
            float bl = (j == NBINS) ? (cw[NBINS] + EPSV) : cw[j];
            idx += (xi >= bl) ? 1 : 0;
        }
        idx = idx < 0 ? 0 : (idx > NBINS - 1 ? NBINS - 1 : idx);

        // register-resident gathers -> v_cndmask chains
        float cw_k = cw[0], cw_k1 = cw[1];
        float ch_k = ch[0], ch_k1 = ch[1];
        float d_k  = dv[0], d_k1  = dv[1];
#pragma unroll
        for (int k = 1; k < NBINS; ++k) {
            bool p = (idx == k);
            cw_k  = p ? cw[k]     : cw_k;
            cw_k1 = p ? cw[k + 1] : cw_k1;
            ch_k  = p ? ch[k]     : ch_k;
            ch_k1 = p ? ch[k + 1] : ch_k1;
            d_k   = p ? dv[k]     : d_k;
            d_k1  = p ? dv[k + 1] : d_k1;
        }

        const float bwd   = cw_k1 - cw_k;
        const float bh    = ch_k1 - ch_k;
        const float theta = (xi - cw_k) / bwd;
        const float omt   = 1.0f - theta;
        const float t1m   = theta * omt;
        const float th2   = theta * theta;
        const float numr  = bh * (d_k * th2 + d_k1 * t1m);
        const float den   = d_k * th2 + 2.0f * d_k1 * t1m + omt * omt;
        const float out_in = ch_k + numr / den;
        const float den2  = den * den;
        const float ld_in  = __logf((d_k1 * d_k * bh * den2) / (bwd * den2));
        const float ld_out = __logf(dv[0]);

        __builtin_nontemporal_store(inside ? out_in : xv, &out[base + tid]);
        lad = inside ? ld_in : ld_out;
    }

    // ---- block reduction of log|det| (wave32 shuffles + LDS) ----
#pragma unroll
    for (int off = 16; off > 0; off >>= 1) lad += __shfl_down(lad, off);
    if ((tid & 31) == 0) sred[tid >> 5] = lad;
    __syncthreads();
    if (tid == 0) {
        float s = 0.0f;
#pragma unroll
        for (int w = 0; w < THREADS / 32; ++w) s += sred[w];
        partial[blockIdx.x] = s;
    }
}

__global__ __launch_bounds__(1024)
void sum_partials(const float* __restrict__ p, int n, float* __restrict__ dst)
{
    __shared__ float sred[32];
    float s = 0.0f;
    for (int i = threadIdx.x; i < n; i += blockDim.x) s += p[i];
#pragma unroll
    for (int off = 16; off > 0; off >>= 1) s += __shfl_down(s, off);
    if ((threadIdx.x & 31) == 0) sred[threadIdx.x >> 5] = s;
    __syncthreads();
    if (threadIdx.x == 0) {
        float t = 0.0f;
        const int nw = (int)((blockDim.x + 31) >> 5);
        for (int w = 0; w < nw; ++w) t += sred[w];
        dst[0] = t;
    }
}

extern "C" void kernel_launch(void* const* d_in, const int* in_sizes, int n_in,
                              void* d_out, int out_size, void* d_ws, size_t ws_size,
                              hipStream_t stream)
{
    const float* x      = (const float*)d_in[0];
    const float* params = (const float*)d_in[1];
    float* out = (float*)d_out;
    const int n = in_sizes[0];
    const int blocks = (n + THREADS - 1) / THREADS;   // 16384 -> needs 64 KB of d_ws
    float* partial = (float*)d_ws;

    rqs_kernel<<<blocks, THREADS, 0, stream>>>(x, params, out, partial, n);
    sum_partials<<<1, 1024, 0, stream>>>(partial, blocks, out + n); // scalar at d_out[N]
}